// GmpnnCSNetDrugBank_47081431499263
// MI455X (gfx1250) — compile-verified
//
#include <hip/hip_runtime.h>
#include <cstddef>

typedef __attribute__((ext_vector_type(2))) float v2f;
typedef __attribute__((ext_vector_type(8))) float v8f;

#define WPB 8  // waves per block (256 threads, wave32)

// ---------------------------------------------------------------------------
// Input index map (setup_inputs dict order; params dict in insertion order)
// ---------------------------------------------------------------------------
enum {
  IN_X = 0, IN_EDGE_FEATS = 1,
  P_MLP_W1 = 2, P_MLP_B1, P_MLP_P1, P_MLP_W2, P_MLP_B2, P_BN1_G, P_BN1_B, P_MLP_P2,
  P_MLP_W3, P_MLP_B3, P_BN2_G, P_BN2_B,                    // 10..13
  P_W_I = 14, P_W_J, P_B_IJ, P_EE_W, P_EE_B, P_SML_P, P_SML_W, P_SML_B, // 14..21
  P_BNL1_G = 22, P_BNL1_B, P_L1_W, P_L1_B,                 // 22..25
  P_BNL2_G = 26, P_BNL2_B, P_L2_P, P_L2_W, P_L2_B,         // 26..30
  P_BNL3_G = 31, P_BNL3_B, P_L3_P, P_L3_W, P_L3_B,         // 31..35
  P_BNL4_G = 36, P_BNL4_B, P_L4_P, P_L4_W, P_L4_B,         // 36..40
  P_I_PRO = 41, P_J_PRO, P_WQ, P_WK, P_CA_B, P_CA_A, P_REL_EMBS, // 41..47
  IN_EDGE_INDEX = 48, IN_LG_EDGE_INDEX = 49, IN_PAIR_EDGE_INDEX = 50,
  IN_PE_BATCH = 51, IN_DPI = 52, IN_NODE_J = 53, IN_NODE_I = 54, IN_RELS = 55
};

// ---------------------------------------------------------------------------
// WMMA FP32 GEMM: C[M,NC] = f(A[M,K]) @ B[K,NC] (+ bias[NC])
//   GATHER: 0 = dense rows, 1 = A[idx0[r]], 2 = A[idx0[r]] + A2[idx1[r]]
//   AFFINE: per-K-column scale/shift (folded BatchNorm) applied to A
//   PRELU : scalar-parameter PReLU applied to A (after affine)
// K and NC are compile-time: K-loop fully unrolls into back-to-back
// v_wmma_f32_16x16x4_f32 chains; A fragment is preloaded/transformed once per
// wave (v2f 8-byte loads) and reused across all NC/16 column tiles.
// A layout (ISA 7.12.2): lane<16 -> K=kk+{0,1}; lane>=16 -> K=kk+{2,3}
// C/D layout: VGPR i -> row i (lanes 0-15) / row i+8 (lanes 16-31)
// ---------------------------------------------------------------------------
template<int GATHER, bool AFFINE, bool PRELU, bool BIAS, int K, int NC>
__global__ __launch_bounds__(256)
void gemm16(const float* __restrict__ A, const float* __restrict__ A2,
            const float* __restrict__ Bm, const float* __restrict__ bias,
            float* __restrict__ C,
            const int* __restrict__ idx0, const int* __restrict__ idx1,
            const float* __restrict__ scale, const float* __restrict__ shift,
            const float* __restrict__ prelu_a, int M)
{
  constexpr int NSTEP = (K + 3) / 4;
  const int lane  = threadIdx.x & 31;
  const int wave  = threadIdx.x >> 5;
  const int mrow  = lane & 15;
  const int khalf = (lane >> 4) & 1;     // 0: K+{0,1}, 1: K+{2,3}
  const int rowTile = blockIdx.x * WPB + wave;
  if (rowTile >= (M >> 4)) return;       // wave-uniform exit: EXEC all-1 for WMMA

  const float pa = PRELU ? prelu_a[0] : 0.0f;
  const int r = rowTile * 16 + mrow;
  const float* arow0;
  const float* arow1 = nullptr;
  if (GATHER == 0)      arow0 = A + (size_t)r * K;
  else if (GATHER == 1) arow0 = A + (size_t)idx0[r] * K;
  else { arow0 = A + (size_t)idx0[r] * K; arow1 = A2 + (size_t)idx1[r] * K; }

  // ---- preload + transform A fragment (reused across all column tiles) ----
  float a0[NSTEP], a1[NSTEP];
  #pragma unroll
  for (int s = 0; s < NSTEP; ++s) {
    const int k0 = s * 4 + 2 * khalf;
    float v0 = 0.f, v1 = 0.f;
    if (k0 + 1 < K) {                     // 8-byte vector load (k0 even, rows 8B-aligned)
      v2f av = *(const v2f*)(arow0 + k0);
      v0 = av[0]; v1 = av[1];
      if (GATHER == 2) { v2f aw = *(const v2f*)(arow1 + k0); v0 += aw[0]; v1 += aw[1]; }
      if (AFFINE) { v0 = v0 * scale[k0] + shift[k0]; v1 = v1 * scale[k0+1] + shift[k0+1]; }
      if (PRELU)  { v0 = (v0 >= 0.f) ? v0 : pa * v0; v1 = (v1 >= 0.f) ? v1 : pa * v1; }
    } else if (k0 < K) {                  // ragged tail (K=70, K=6)
      v0 = arow0[k0];
      if (GATHER == 2) v0 += arow1[k0];
      if (AFFINE) v0 = v0 * scale[k0] + shift[k0];
      if (PRELU)  v0 = (v0 >= 0.f) ? v0 : pa * v0;
    }
    a0[s] = v0; a1[s] = v1;
  }

  #pragma unroll
  for (int nt = 0; nt < NC / 16; ++nt) {
    const int col = nt * 16 + mrow;
    v8f acc = {0.f, 0.f, 0.f, 0.f, 0.f, 0.f, 0.f, 0.f};
    #pragma unroll
    for (int s = 0; s < NSTEP; ++s) {
      const int k0 = s * 4 + 2 * khalf;
      v2f a, b;
      a[0] = a0[s]; a[1] = a1[s];
      b[0] = (k0     < K) ? Bm[(size_t)k0       * NC + col] : 0.f;
      b[1] = (k0 + 1 < K) ? Bm[(size_t)(k0 + 1) * NC + col] : 0.f;
      acc = __builtin_amdgcn_wmma_f32_16x16x4_f32(false, a, false, b, (short)0, acc,
                                                  false, false);
    }
    const float bc = BIAS ? bias[col] : 0.f;
    #pragma unroll
    for (int i = 0; i < 8; ++i) {
      const int orow = rowTile * 16 + i + 8 * khalf;
      C[(size_t)orow * NC + col] = acc[i] + bc;
    }
  }
}

// ---------------------------------------------------------------------------
// Utility / elementwise kernels
// ---------------------------------------------------------------------------
__global__ void k_zero(float* __restrict__ p, size_t n) {
  size_t i = (size_t)blockIdx.x * blockDim.x + threadIdx.x;
  if (i < n) p[i] = 0.f;
}

__global__ void k_copy(const float* __restrict__ a, float* __restrict__ b, size_t n) {
  size_t i = (size_t)blockIdx.x * blockDim.x + threadIdx.x;
  if (i < n) b[i] = a[i];
}

// y = (a + b) * 0.5   (y may alias b)
__global__ void k_avg(const float* __restrict__ a, const float* __restrict__ b,
                      float* __restrict__ y, size_t n) {
  size_t i = (size_t)blockIdx.x * blockDim.x + threadIdx.x;
  if (i < n) y[i] = (a[i] + b[i]) * 0.5f;
}

// y[i] = x[i]*scale[c] + shift[c],  c = i & cmask
__global__ void k_affine(const float* __restrict__ x, const float* __restrict__ scale,
                         const float* __restrict__ shift, float* __restrict__ y,
                         size_t n, int cmask) {
  size_t i = (size_t)blockIdx.x * blockDim.x + threadIdx.x;
  if (i < n) { int c = (int)(i & (size_t)cmask); y[i] = x[i] * scale[c] + shift[c]; }
}

// ---------------------------------------------------------------------------
// BatchNorm batch statistics: two-stage (LDS partial -> global atomics)
// ---------------------------------------------------------------------------
__global__ __launch_bounds__(256)
void k_stats_partial(const float* __restrict__ X, float* __restrict__ gsum,
                     float* __restrict__ gsq, size_t total, int cmask, int C) {
  __shared__ float ls[128], lq[128];
  int tid = threadIdx.x;
  for (int i = tid; i < C; i += blockDim.x) { ls[i] = 0.f; lq[i] = 0.f; }
  __syncthreads();
  size_t stride = (size_t)gridDim.x * blockDim.x;
  for (size_t i = (size_t)blockIdx.x * blockDim.x + tid; i < total; i += stride) {
    float v = X[i];
    int c = (int)(i & (size_t)cmask);
    atomicAdd(&ls[c], v);
    atomicAdd(&lq[c], v * v);
  }
  __syncthreads();
  for (int i = tid; i < C; i += blockDim.x) {
    atomicAdd(&gsum[i], ls[i]);
    atomicAdd(&gsq[i], lq[i]);
  }
}

__global__ void k_stats_finalize(const float* __restrict__ gsum, const float* __restrict__ gsq,
                                 const float* __restrict__ g, const float* __restrict__ b,
                                 float* __restrict__ scale, float* __restrict__ shift,
                                 int Nrows, int C) {
  int c = blockIdx.x * blockDim.x + threadIdx.x;
  if (c < C) {
    float inv = 1.f / (float)Nrows;
    float m = gsum[c] * inv;
    float var = gsq[c] * inv - m * m;
    float s = g[c] * rsqrtf(var + 1e-5f);
    scale[c] = s;
    shift[c] = b[c] - m * s;
  }
}

// ---------------------------------------------------------------------------
// Graph kernels
// ---------------------------------------------------------------------------
__global__ void k_degree(const int* __restrict__ dst, float* __restrict__ deg, int E) {
  int e = blockIdx.x * blockDim.x + threadIdx.x;
  if (e < E) atomicAdd(&deg[dst[e]], 1.0f);
}

// ew[e] = sigmoid( dot64(sml[e], ef[e]) / deg[src[e]] )  -- one wave per edge
__global__ __launch_bounds__(256)
void k_edge_gate(const float* __restrict__ sml, const float* __restrict__ ef,
                 const int* __restrict__ src, const float* __restrict__ deg,
                 float* __restrict__ ew, int E) {
  int lane = threadIdx.x & 31;
  int e = blockIdx.x * (blockDim.x >> 5) + (threadIdx.x >> 5);
  if (e >= E) return;
  const float* sr = sml + (size_t)e * 64;
  const float* er = ef + (size_t)e * 64;
  float p = sr[lane] * er[lane] + sr[lane + 32] * er[lane + 32];
  for (int off = 16; off; off >>= 1) p += __shfl_xor(p, off);
  if (lane == 0) {
    float a = p / deg[src[e]];
    ew[e] = 1.f / (1.f + expf(-a));
  }
}

// edge_attr = h[src]*ew ; out = edge_attr
__global__ void k_edge_attr(const float* __restrict__ h, const int* __restrict__ src,
                            const float* __restrict__ ew, float* __restrict__ eattr,
                            float* __restrict__ out, size_t total) {
  size_t i = (size_t)blockIdx.x * blockDim.x + threadIdx.x;
  if (i < total) {
    size_t e = i >> 6; int c = (int)(i & 63);
    float v = h[(size_t)src[e] * 64 + c] * ew[e];
    eattr[i] = v; out[i] = v;
  }
}

// msg[lg1[l]] += out[lg0[l]]
__global__ void k_lg_scatter(const float* __restrict__ out, const int* __restrict__ lg0,
                             const int* __restrict__ lg1, float* __restrict__ msg,
                             size_t total) {
  size_t i = (size_t)blockIdx.x * blockDim.x + threadIdx.x;
  if (i < total) {
    size_t l = i >> 6; int c = (int)(i & 63);
    atomicAdd(&msg[(size_t)lg1[l] * 64 + c], out[(size_t)lg0[l] * 64 + c]);
  }
}

// out = edge_attr + msg*ew
__global__ void k_lg_combine(const float* __restrict__ eattr, const float* __restrict__ msg,
                             const float* __restrict__ ew, float* __restrict__ out,
                             size_t total) {
  size_t i = (size_t)blockIdx.x * blockDim.x + threadIdx.x;
  if (i < total) { size_t e = i >> 6; out[i] = eattr[i] + msg[i] * ew[e]; }
}

// agg[dst[e]] += out[e]
__global__ void k_agg_scatter(const float* __restrict__ out, const int* __restrict__ dst,
                              float* __restrict__ agg, size_t total) {
  size_t i = (size_t)blockIdx.x * blockDim.x + threadIdx.x;
  if (i < total) {
    size_t e = i >> 6; int c = (int)(i & 63);
    atomicAdd(&agg[(size_t)dst[e] * 64 + c], out[i]);
  }
}

// pair co-attention: one wave per pair-edge
__global__ __launch_bounds__(256)
void k_pair(const float* __restrict__ q, const float* __restrict__ kmat,
            const float* __restrict__ ipro, const float* __restrict__ jpro,
            const float* __restrict__ ca_b, const float* __restrict__ ca_a,
            const int* __restrict__ pe0, const int* __restrict__ pe1,
            const int* __restrict__ pbatch, float* __restrict__ pairsum, int P) {
  int lane = threadIdx.x & 31;
  int p = blockIdx.x * (blockDim.x >> 5) + (threadIdx.x >> 5);
  if (p >= P) return;
  int i0 = pe0[p], i1 = pe1[p];
  const float* qr = q + (size_t)i1 * 64;
  const float* kr = kmat + (size_t)i0 * 64;
  int c0 = lane, c1 = lane + 32;
  float e0 = tanhf(qr[c0] + kr[c0] + ca_b[c0]);
  float e1 = tanhf(qr[c1] + kr[c1] + ca_b[c1]);
  float part = e0 * ca_a[c0] + e1 * ca_a[c1];
  for (int off = 16; off; off >>= 1) part += __shfl_xor(part, off);  // all lanes = attn
  const float* ir = ipro + (size_t)i1 * 64;
  const float* jr = jpro + (size_t)i0 * 64;
  float* dp = pairsum + (size_t)pbatch[p] * 64;
  atomicAdd(&dp[c0], part * ir[c0] * jr[c0]);
  atomicAdd(&dp[c1], part * ir[c1] * jr[c1]);
}

// scores[t] = dot64(pairsum[dpi[t]], rel_embs[rels_full[t]])
__global__ __launch_bounds__(256)
void k_score(const float* __restrict__ pairsum, const float* __restrict__ rel_embs,
             const int* __restrict__ dpi, const int* __restrict__ rels,
             float* __restrict__ outp, int T, int B, int neg_n) {
  int lane = threadIdx.x & 31;
  int t = blockIdx.x * (blockDim.x >> 5) + (threadIdx.x >> 5);
  if (t >= T) return;
  int rel = (t < B) ? rels[t] : rels[(t - B) / neg_n];
  const float* pr = pairsum + (size_t)dpi[t] * 64;
  const float* re = rel_embs + (size_t)rel * 64;
  float p = pr[lane] * re[lane] + pr[lane + 32] * re[lane + 32];
  for (int off = 16; off; off >>= 1) p += __shfl_xor(p, off);
  if (lane == 0) outp[t] = p;
}

// ---------------------------------------------------------------------------
// Host orchestration
// ---------------------------------------------------------------------------
static inline unsigned cdiv(size_t n, unsigned t) { return (unsigned)((n + t - 1) / t); }

extern "C" void kernel_launch(void* const* d_in, const int* in_sizes, int n_in,
                              void* d_out, int out_size, void* d_ws, size_t ws_size,
                              hipStream_t stream) {
  constexpr int IN = 70, H = 64, S = 128, EF = 6, U = 1024, NITER = 4;
  const int N   = in_sizes[IN_X] / IN;           // 100000
  const int E   = in_sizes[IN_EDGE_FEATS] / EF;  // 400000
  const int ELG = in_sizes[IN_LG_EDGE_INDEX] / 2;// 1200000
  const int P   = in_sizes[IN_PE_BATCH];         // 640000
  const int Mp  = in_sizes[IN_NODE_J];           // 50000
  const int T   = in_sizes[IN_DPI];              // 1024
  const int Bsz = in_sizes[IN_RELS];             // 512
  const int neg_n = (T - Bsz) / Bsz;
  (void)n_in; (void)out_size; (void)ws_size;

  #define FP(i) ((const float*)d_in[i])
  #define IP(i) ((const int*)d_in[i])

  const int* e_src = IP(IN_EDGE_INDEX);
  const int* e_dst = e_src + E;
  const int* lg0 = IP(IN_LG_EDGE_INDEX);
  const int* lg1 = lg0 + ELG;
  const int* pe0 = IP(IN_PAIR_EDGE_INDEX);
  const int* pe1 = pe0 + P;

  // ----- workspace layout (float elements) -----
  float* ws = (float*)d_ws;
  const size_t E64 = (size_t)E * 64, N64 = (size_t)N * 64;
  const size_t N128 = (size_t)N * 128, M64 = (size_t)Mp * 64;
  float* e0 = ws;                 // ef -> edge_attr -> h128a(+h128b)
  float* e1 = e0 + E64;           // smlout -> out -> h128c
  float* e2 = e1 + E64;           // msg -> keys/queries/ipro/jpro
  float* n0 = e2 + E64;           // tmp0 -> tmp2 -> agg
  float* n1 = n0 + N64;           // tmp1 -> hwj
  float* n2 = n1 + N64;           // h
  float* n3 = n2 + N64;           // hwi (+b_ij)
  float* ew = n3 + N64;           // [E]
  float* deg = ew + E;            // [N]
  float* pairsum = deg + N;       // [U*64]
  float* ssum = pairsum + (size_t)U * 64;  // [128]
  float* ssq  = ssum + 128;                // [128]
  float* scA  = ssq + 128;                 // [128] folded BN scale
  float* shA  = scA + 128;                 // [128] folded BN shift

  float* h128a = e0;              // [N,128]
  float* h128b = e0 + N128;       // [N,128]  (fits: E64 == 2*N128)
  float* h128c = e1;              // [N,128]
  float* keys    = e2;
  float* queries = e2 + M64;
  float* ipro    = e2 + 2 * M64;
  float* jpro    = e2 + 3 * M64;

  const unsigned TB = 256;
  const unsigned gN16 = cdiv((size_t)(N / 16), WPB);
  const unsigned gE16 = cdiv((size_t)(E / 16), WPB);
  const unsigned gM16 = cdiv((size_t)(Mp / 16), WPB);

  auto stats = [&](const float* X, int rows, int C, const float* g, const float* b) {
    k_zero<<<1, 256, 0, stream>>>(ssum, 256);  // ssum+ssq contiguous
    k_stats_partial<<<512, TB, 0, stream>>>(X, ssum, ssq, (size_t)rows * C, C - 1, C);
    k_stats_finalize<<<1, 128, 0, stream>>>(ssum, ssq, g, b, scA, shA, rows, C);
  };

  // ===== top MLP =====
  gemm16<0,false,false,true,IN,H><<<gN16, TB, 0, stream>>>(FP(IN_X), nullptr, FP(P_MLP_W1),
      FP(P_MLP_B1), n0, nullptr, nullptr, nullptr, nullptr, nullptr, N);
  gemm16<0,false,true,true,H,H><<<gN16, TB, 0, stream>>>(n0, nullptr, FP(P_MLP_W2),
      FP(P_MLP_B2), n1, nullptr, nullptr, nullptr, nullptr, FP(P_MLP_P1), N);
  stats(n1, N, H, FP(P_BN1_G), FP(P_BN1_B));
  gemm16<0,true,true,true,H,H><<<gN16, TB, 0, stream>>>(n1, nullptr, FP(P_MLP_W3),
      FP(P_MLP_B3), n0, nullptr, nullptr, scA, shA, FP(P_MLP_P2), N);
  stats(n0, N, H, FP(P_BN2_G), FP(P_BN2_B));
  k_affine<<<cdiv(N64, TB), TB, 0, stream>>>(n0, scA, shA, n2, N64, 63);  // h

  // ===== GmpnnBlock =====
  gemm16<0,false,false,true,H,H><<<gN16, TB, 0, stream>>>(n2, nullptr, FP(P_W_I),
      FP(P_B_IJ), n3, nullptr, nullptr, nullptr, nullptr, nullptr, N);   // hwi + b_ij
  gemm16<0,false,false,false,H,H><<<gN16, TB, 0, stream>>>(n2, nullptr, FP(P_W_J),
      nullptr, n1, nullptr, nullptr, nullptr, nullptr, nullptr, N);      // hwj
  gemm16<0,false,false,true,EF,H><<<gE16, TB, 0, stream>>>(FP(IN_EDGE_FEATS), nullptr,
      FP(P_EE_W), FP(P_EE_B), e0, nullptr, nullptr, nullptr, nullptr, nullptr, E);
  k_zero<<<cdiv((size_t)N, TB), TB, 0, stream>>>(deg, (size_t)N);
  k_degree<<<cdiv((size_t)E, TB), TB, 0, stream>>>(e_dst, deg, E);
  // smlout = prelu(hwi[dst]+hwj[src], sml_p) @ sml_w + sml_b
  gemm16<2,false,true,true,H,H><<<gE16, TB, 0, stream>>>(n3, n1, FP(P_SML_W), FP(P_SML_B),
      e1, e_dst, e_src, nullptr, nullptr, FP(P_SML_P), E);
  k_edge_gate<<<cdiv((size_t)E, WPB), TB, 0, stream>>>(e1, e0, e_src, deg, ew, E);
  k_edge_attr<<<cdiv(E64, TB), TB, 0, stream>>>(n2, e_src, ew, e0, e1, E64);
  const size_t LG64 = (size_t)ELG * 64;
  for (int it = 0; it < NITER; ++it) {
    k_zero<<<cdiv(E64, TB), TB, 0, stream>>>(e2, E64);
    k_lg_scatter<<<cdiv(LG64, TB), TB, 0, stream>>>(e1, lg0, lg1, e2, LG64);
    k_lg_combine<<<cdiv(E64, TB), TB, 0, stream>>>(e0, e2, ew, e1, E64);
  }
  k_copy<<<cdiv(N64, TB), TB, 0, stream>>>(n2, n0, N64);            // agg = h
  k_agg_scatter<<<cdiv(E64, TB), TB, 0, stream>>>(e1, e_dst, n0, E64);

  // ===== block MLP (S=128) =====
  stats(n0, N, H, FP(P_BNL1_G), FP(P_BNL1_B));
  gemm16<0,true,false,true,H,S><<<gN16, TB, 0, stream>>>(n0, nullptr, FP(P_L1_W), FP(P_L1_B),
      h128a, nullptr, nullptr, scA, shA, nullptr, N);
  stats(h128a, N, S, FP(P_BNL2_G), FP(P_BNL2_B));
  gemm16<0,true,true,true,S,S><<<gN16, TB, 0, stream>>>(h128a, nullptr, FP(P_L2_W), FP(P_L2_B),
      h128b, nullptr, nullptr, scA, shA, FP(P_L2_P), N);
  stats(h128b, N, S, FP(P_BNL3_G), FP(P_BNL3_B));
  gemm16<0,true,true,true,S,S><<<gN16, TB, 0, stream>>>(h128b, nullptr, FP(P_L3_W), FP(P_L3_B),
      h128c, nullptr, nullptr, scA, shA, FP(P_L3_P), N);
  k_avg<<<cdiv(N128, TB), TB, 0, stream>>>(h128c, h128a, h128a, N128);
  stats(h128a, N, S, FP(P_BNL4_G), FP(P_BNL4_B));
  gemm16<0,true,true,true,S,S><<<gN16, TB, 0, stream>>>(h128a, nullptr, FP(P_L4_W), FP(P_L4_B),
      h128b, nullptr, nullptr, scA, shA, FP(P_L4_P), N);
  k_avg<<<cdiv(N128, TB), TB, 0, stream>>>(h128b, h128a, h128a, N128);

  // ===== pair co-attention =====
  gemm16<1,false,false,false,S,H><<<gM16, TB, 0, stream>>>(h128a, nullptr, FP(P_WK), nullptr,
      keys, IP(IN_NODE_J), nullptr, nullptr, nullptr, nullptr, Mp);
  gemm16<1,false,false,false,S,H><<<gM16, TB, 0, stream>>>(h128a, nullptr, FP(P_WQ), nullptr,
      queries, IP(IN_NODE_I), nullptr, nullptr, nullptr, nullptr, Mp);
  gemm16<1,false,false,false,S,H><<<gM16, TB, 0, stream>>>(h128a, nullptr, FP(P_I_PRO), nullptr,
      ipro, IP(IN_NODE_I), nullptr, nullptr, nullptr, nullptr, Mp);
  gemm16<1,false,false,false,S,H><<<gM16, TB, 0, stream>>>(h128a, nullptr, FP(P_J_PRO), nullptr,
      jpro, IP(IN_NODE_J), nullptr, nullptr, nullptr, nullptr, Mp);
  k_zero<<<cdiv((size_t)U * 64, TB), TB, 0, stream>>>(pairsum, (size_t)U * 64);
  k_pair<<<cdiv((size_t)P, WPB), TB, 0, stream>>>(queries, keys, ipro, jpro,
      FP(P_CA_B), FP(P_CA_A), pe0, pe1, IP(IN_PE_BATCH), pairsum, P);

  // ===== score =====
  k_score<<<cdiv((size_t)T, WPB), TB, 0, stream>>>(pairsum, FP(P_REL_EMBS),
      IP(IN_DPI), IP(IN_RELS), (float*)d_out, T, Bsz, neg_n);
  #undef FP
  #undef IP
}